// ParametricKernel_16844861735005
// MI455X (gfx1250) — compile-verified
//
#include <hip/hip_runtime.h>
#include <hip/hip_bf16.h>

typedef float v2f __attribute__((ext_vector_type(2)));
typedef float v8f __attribute__((ext_vector_type(8)));

#define LOG2E 1.4426950408889634f

__device__ __forceinline__ float rl(float v, int l) {
    return __builtin_bit_cast(float,
        __builtin_amdgcn_readlane(__builtin_bit_cast(int, v), l));
}

// raw v_exp_f32: no libm denormal fixups; args < -126 underflow to 0, which is
// the desired limit for exp(-0.5*d2) at large distances.
__device__ __forceinline__ float fast_exp(float x) {
    return __builtin_amdgcn_exp2f(x * LOG2E);
}

// One wave32 per batch element. B = 65536, M = 32, D = 2.
// out layout: [0, B*32) mean_factors, [B*32, B*33) variances.
__global__ __launch_bounds__(256) void vecchia_wmma_kernel(
    const float* __restrict__ locs,        // [N,2]
    const float* __restrict__ log_ls,      // [1]
    const float* __restrict__ log_sigmasq, // [1]
    const int*   __restrict__ batch_idx,   // [B]
    const int*   __restrict__ cond,        // [N,32]
    float* __restrict__ out_mean,          // [B,32]
    float* __restrict__ out_var,           // [B]
    int B)
{
    const int lane = threadIdx.x & 31;
    const int wave = threadIdx.x >> 5;
    const int b    = blockIdx.x * 8 + wave;
    if (b >= B) return;                    // wave-uniform exit: EXEC stays full

    const float inv_ls  = fast_exp(-log_ls[0]);          // 1/exp(log_ls)
    const float sigmasq = fast_exp(log_sigmasq[0]);

    const int idx = batch_idx[b];
    const int nn  = cond[idx * 32 + lane];
    const float validf = (nn >= 0) ? 1.0f : 0.0f;
    const int   nnc    = (nn >= 0) ? nn : 0;

    const float2 pl = ((const float2*)locs)[nnc];
    const float x  = pl.x * inv_ls;
    const float y  = pl.y * inv_ls;
    const float nh = -0.5f * (x * x + y * y);             // -0.5*||p||^2

    const float2 cl = ((const float2*)locs)[idx];
    const float cx = cl.x * inv_ls, cy = cl.y * inv_ls;
    const float dx = cx - x, dy = cy - y;
    const float kc = validf * fast_exp(-0.5f * (dx * dx + dy * dy));

    // ---- Gram logits via V_WMMA_F32_16X16X4_F32 (rank-4 augmented) ----
    // A row i = (x_i, y_i, nh_i, 1); B col j = (x_j, y_j, 1, nh_j)
    // => D[i][j] = x_i x_j + y_i y_j + nh_i + nh_j = -0.5 * d2(i,j)
    const float xs = __shfl_xor(x, 16, 32);
    const float ys = __shfl_xor(y, 16, 32);
    const float ns = __shfl_xor(nh, 16, 32);
    const bool  lo = lane < 16;

    // A 16x4 f32: lanes 0-15 hold K=0,1 (M=lane); lanes 16-31 hold K=2,3
    v2f A0, A1, Bm0, Bm1;
    A0.x  = lo ? x  : ns;   A0.y  = lo ? y  : 1.0f;   // rows  0..15
    A1.x  = lo ? xs : nh;   A1.y  = lo ? ys : 1.0f;   // rows 16..31
    // B 4x16 f32: lanes 0-15 hold K=0,1 (N=lane); lanes 16-31 hold K=2,3
    Bm0.x = lo ? x  : 1.0f; Bm0.y = lo ? y  : ns;     // cols  0..15
    Bm1.x = lo ? xs : 1.0f; Bm1.y = lo ? ys : nh;     // cols 16..31

    v8f cz = {0.f,0.f,0.f,0.f,0.f,0.f,0.f,0.f};
    v8f D00 = __builtin_amdgcn_wmma_f32_16x16x4_f32(false, A0, false, Bm0, (short)0, cz, false, false);
    v8f D01 = __builtin_amdgcn_wmma_f32_16x16x4_f32(false, A0, false, Bm1, (short)0, cz, false, false);
    v8f D10 = __builtin_amdgcn_wmma_f32_16x16x4_f32(false, A1, false, Bm0, (short)0, cz, false, false);
    v8f D11 = __builtin_amdgcn_wmma_f32_16x16x4_f32(false, A1, false, Bm1, (short)0, cz, false, false);

    // ---- re-tile D (VGPR r = rows r / r+8, lane halves = row halves)
    //      into column-per-lane: a[i] = G[i][lane] ----
    float a[32];
#pragma unroll
    for (int r = 0; r < 8; ++r) {
        float t1 = __shfl_xor(D01[r], 16, 32);
        a[r]      = lo ? D00[r] : t1;
        float t0 = __shfl_xor(D00[r], 16, 32);
        a[r + 8]  = lo ? t0 : D01[r];
        float u1 = __shfl_xor(D11[r], 16, 32);
        a[16 + r] = lo ? D10[r] : u1;
        float u0 = __shfl_xor(D10[r], 16, 32);
        a[24 + r] = lo ? u0 : D11[r];
    }

    // ---- exp, mask (row & col indicators), unit diagonal ----
#pragma unroll
    for (int i = 0; i < 32; ++i) {
        float rv = rl(validf, i);                       // row-i indicator (uniform)
        float e  = fast_exp(a[i]) * validf * rv;
        a[i] = (lane == i) ? 1.0f : e;
    }

    // ---- lane-parallel Cholesky (column per lane), fully unrolled ----
    float rowL[32];                                     // L[lane][k], built for free
    float rdiag = 0.0f, ljk = 0.0f;
#pragma unroll
    for (int k = 0; k < 32; ++k) {
        float pivot = rl(a[k], k);
        float rs    = __frsqrt_rn(pivot);
        rdiag = (lane == k) ? rs : rdiag;               // 1/L[lane][lane]
        float s = (lane == k) ? rs : 1.0f;
#pragma unroll
        for (int i = k; i < 32; ++i) a[i] *= s;         // finalize column k
        float um = (lane > k) ? 1.0f : 0.0f;            // only trailing lanes update
#pragma unroll
        for (int i = k; i < 32; ++i) {
            float Lik = rl(a[i], k);                    // broadcast L[i][k]
            ljk = (lane == i) ? Lik : ljk;              // capture L[lane][k]
            a[i] = fmaf(-Lik, um * ljk, a[i]);          // rank-1 update
        }
        rowL[k] = ljk;
    }

    // ---- forward solve L u = kc (rows are local via rowL) ----
    float u = 0.0f, acc = 0.0f;
#pragma unroll
    for (int j = 0; j < 32; ++j) {
        float t  = (kc - acc) * rdiag;                  // valid at lane j
        float yj = rl(t, j);
        u   = (lane == j) ? yj : u;
        acc = fmaf(rowL[j], yj, acc);                   // acc_p += L[p][j]*y_j
    }

    // ---- backward solve L^T z = u (columns are local via a[]) ----
    float z = 0.0f; acc = 0.0f;
#pragma unroll
    for (int i = 31; i >= 0; --i) {
        float t  = (u - acc) * rdiag;                   // valid at lane i
        float zi = rl(t, i);
        z   = (lane == i) ? zi : z;
        acc = fmaf(a[i], zi, acc);                      // acc_p += L[i][p]*z_i
    }

    // ---- variance: 1 - z . kc ----
    float t = z * kc;
#pragma unroll
    for (int off = 16; off > 0; off >>= 1) t += __shfl_xor(t, off, 32);

    out_mean[(size_t)b * 32 + lane] = z;
    if (lane == 0)
        out_var[b] = fmaxf(sigmasq * (1.0f - t), 1e-9f);
}

extern "C" void kernel_launch(void* const* d_in, const int* in_sizes, int n_in,
                              void* d_out, int out_size, void* d_ws, size_t ws_size,
                              hipStream_t stream) {
    const float* locs        = (const float*)d_in[0];
    const float* log_ls      = (const float*)d_in[1];
    const float* log_sigmasq = (const float*)d_in[2];
    const int*   batch_idx   = (const int*)d_in[3];
    const int*   cond        = (const int*)d_in[4];
    const int B = in_sizes[3];

    float* out      = (float*)d_out;
    float* out_mean = out;
    float* out_var  = out + (size_t)B * 32;

    const int waves_per_block = 8;                      // 256 threads = 8 wave32
    const int grid = (B + waves_per_block - 1) / waves_per_block;
    vecchia_wmma_kernel<<<grid, 256, 0, stream>>>(
        locs, log_ls, log_sigmasq, batch_idx, cond, out_mean, out_var, B);
}